// CVRP_Encoder_27041114096009
// MI455X (gfx1250) — compile-verified
//
#include <hip/hip_runtime.h>
#include <hip/hip_bf16.h>
#include <math.h>

typedef float v2f __attribute__((ext_vector_type(2)));
typedef float v8f __attribute__((ext_vector_type(8)));

#define BB   16
#define NNODE 500
#define DD   128
#define HH   8
#define DKK  16
#define FFF  512
#define LLAY 6
#define EE   8
#define NTOK 502   // valid tokens per batch
#define NPAD 512   // padded tokens per batch

__device__ __forceinline__ v8f wmma4(v2f a, v2f b, v8f c) {
  // D = A(16x4 f32) * B(4x16 f32) + C(16x16 f32)
  return __builtin_amdgcn_wmma_f32_16x16x4_f32(false, a, false, b, (short)0, c, false, false);
}

__device__ __forceinline__ float gelu_exact(float x) {
  return 0.5f * x * (1.0f + erff(x * 0.70710678118654752440f));
}

// ---------------------------------------------------------------- utilities
__global__ void zero_kernel(float* __restrict__ p, size_t n) {
  size_t i = (size_t)blockIdx.x * blockDim.x + threadIdx.x;
  if (i < n) p[i] = 0.0f;
}

// ep = pref @ Wp + bp ; write X[b,501,:] and the ep output slot
__global__ void ep_kernel(const float* __restrict__ pref, const float* __restrict__ Wp,
                          const float* __restrict__ bp, float* __restrict__ X,
                          float* __restrict__ out_ep) {
  int b = blockIdx.x, d = threadIdx.x;
  float v = pref[b * 2 + 0] * Wp[0 * DD + d] + pref[b * 2 + 1] * Wp[1 * DD + d] + bp[d];
  X[((size_t)b * NPAD + (NTOK - 1)) * DD + d] = v;
  out_ep[b * DD + d] = v;
}

// MoE (top-2): one block per token, thread = output feature
__global__ void moe_kernel(const float* __restrict__ xin, const float* __restrict__ Wg,
                           const float* __restrict__ We, const float* __restrict__ be,
                           float* __restrict__ X, float* __restrict__ imp,
                           float* __restrict__ ld, int din, int per_batch, int row_off) {
  int t = blockIdx.x, d = threadIdx.x;
  float xi[3];
  for (int i = 0; i < din; ++i) xi[i] = xin[t * din + i];
  float logit[EE];
  for (int e = 0; e < EE; ++e) {
    float a = 0.0f;
    for (int i = 0; i < din; ++i) a += xi[i] * Wg[i * EE + e];
    logit[e] = a;
  }
  int e0 = 0; float v0 = logit[0];
  for (int e = 1; e < EE; ++e) if (logit[e] > v0) { v0 = logit[e]; e0 = e; }
  int e1 = (e0 == 0) ? 1 : 0; float v1 = logit[e1];
  for (int e = 0; e < EE; ++e)
    if (e != e0 && e != e1 && logit[e] > v1) { v1 = logit[e]; e1 = e; }
  // softmax over (v0 >= v1)
  float ex = __expf(v1 - v0);
  float inv = 1.0f / (1.0f + ex);
  float g0 = inv, g1 = ex * inv;

  float a0 = 0.0f, a1 = 0.0f;
  const float* W0 = We + (size_t)e0 * din * DD;
  const float* W1 = We + (size_t)e1 * din * DD;
  for (int i = 0; i < din; ++i) {
    a0 += xi[i] * W0[i * DD + d];
    a1 += xi[i] * W1[i * DD + d];
  }
  float acc = g0 * (a0 + be[e0 * DD + d]) + g1 * (a1 + be[e1 * DD + d]);
  size_t row = (size_t)(t / per_batch) * NPAD + row_off + (t % per_batch);
  X[row * DD + d] = acc;
  if (d == 0) {
    atomicAdd(&imp[e0], g0); atomicAdd(&imp[e1], g1);
    atomicAdd(&ld[e0], 1.0f); atomicAdd(&ld[e1], 1.0f);
  }
}

// loss = sum of 4 cv^2 (ddof=1) over [imp_d, ld_d, imp_n, ld_n]
__global__ void loss_kernel(const float* __restrict__ st, float* __restrict__ out) {
  if (threadIdx.x == 0) {
    float total = 0.0f;
    for (int v = 0; v < 4; ++v) {
      const float* x = st + v * EE;
      float s = 0.0f;
      for (int e = 0; e < EE; ++e) s += x[e];
      float mu = s / EE;
      float ss = 0.0f;
      for (int e = 0; e < EE; ++e) { float dd = x[e] - mu; ss += dd * dd; }
      float var = ss / (EE - 1);
      total += var / (mu * mu + 1e-10f);
    }
    out[0] = total;
  }
}

// ---------------------------------------------------------------- WMMA GEMM
// C(MxN) = A(MxK) @ W(KxN) [+ bias] [gelu]; row-major, M%128==0, N%64==0, K%16==0
// Block tile 128x64, 4 waves; each wave computes 2 (M) x 4 (N) WMMA tiles.
__global__ __launch_bounds__(128)
void gemm_wmma(const float* __restrict__ A, const float* __restrict__ W,
               const float* __restrict__ bias, float* __restrict__ C,
               int M, int K, int N, int act) {
  __shared__ float sA[128][17];
  __shared__ float sW[16][65];
  int tid = threadIdx.x;
  int wave = tid >> 5;
  int lane = tid & 31;
  int lo = lane & 15;
  int hi = lane >> 4;
  int m0 = blockIdx.x * 128;
  int n0 = blockIdx.y * 64;

  v8f acc[2][4];
#pragma unroll
  for (int ms = 0; ms < 2; ++ms)
#pragma unroll
    for (int t = 0; t < 4; ++t)
#pragma unroll
      for (int e = 0; e < 8; ++e) acc[ms][t][e] = 0.0f;

  int kchunks = K >> 4;
  for (int kc = 0; kc < kchunks; ++kc) {
    int k0 = kc << 4;
    __syncthreads();
#pragma unroll
    for (int it = 0; it < 4; ++it) {  // A tile 128x16 = 512 float4
      int idx = tid * 4 + it;
      int row = idx >> 2, colv = (idx & 3) * 4;
      float4 v = *reinterpret_cast<const float4*>(A + (size_t)(m0 + row) * K + k0 + colv);
      sA[row][colv + 0] = v.x; sA[row][colv + 1] = v.y;
      sA[row][colv + 2] = v.z; sA[row][colv + 3] = v.w;
    }
#pragma unroll
    for (int it = 0; it < 2; ++it) {  // W tile 16x64 = 256 float4
      int idx = tid * 2 + it;
      int row = idx >> 4, colv = (idx & 15) * 4;
      float4 v = *reinterpret_cast<const float4*>(W + (size_t)(k0 + row) * N + n0 + colv);
      sW[row][colv + 0] = v.x; sW[row][colv + 1] = v.y;
      sW[row][colv + 2] = v.z; sW[row][colv + 3] = v.w;
    }
    __syncthreads();
#pragma unroll
    for (int kk = 0; kk < 4; ++kk) {
      v2f af[2];
#pragma unroll
      for (int ms = 0; ms < 2; ++ms) {
        af[ms][0] = sA[wave * 32 + ms * 16 + lo][kk * 4 + 2 * hi + 0];
        af[ms][1] = sA[wave * 32 + ms * 16 + lo][kk * 4 + 2 * hi + 1];
      }
#pragma unroll
      for (int t = 0; t < 4; ++t) {
        v2f bf;
        bf[0] = sW[kk * 4 + 2 * hi + 0][t * 16 + lo];
        bf[1] = sW[kk * 4 + 2 * hi + 1][t * 16 + lo];
#pragma unroll
        for (int ms = 0; ms < 2; ++ms)
          acc[ms][t] = wmma4(af[ms], bf, acc[ms][t]);
      }
    }
  }
#pragma unroll
  for (int ms = 0; ms < 2; ++ms) {
#pragma unroll
    for (int t = 0; t < 4; ++t) {
      int col = n0 + t * 16 + lo;
      float bv = bias ? bias[col] : 0.0f;
#pragma unroll
      for (int r = 0; r < 8; ++r) {
        int row = m0 + wave * 32 + ms * 16 + r + 8 * hi;
        float v = acc[ms][t][r] + bv;
        if (act == 1) v = gelu_exact(v);
        C[(size_t)row * N + col] = v;
      }
    }
  }
}

// ---------------------------------------------------------------- attention
// vpref[b,:] = X[b,501,:] @ Wv2  (cross-attn over a single key == broadcast V)
__global__ void prefv_kernel(const float* __restrict__ X, const float* __restrict__ Wv2,
                             float* __restrict__ vp) {
  int b = blockIdx.x, d = threadIdx.x;
  const float* x = X + ((size_t)b * NPAD + (NTOK - 1)) * DD;
  float a = 0.0f;
  for (int k = 0; k < DD; ++k) a += x[k] * Wv2[k * DD + d];
  vp[b * DD + d] = a;
}

// one wave per (query-tile, head, batch); flash softmax over 502 keys
__global__ __launch_bounds__(32)
void attn_kernel(const float* __restrict__ Q, const float* __restrict__ Kb,
                 const float* __restrict__ Vb, const float* __restrict__ vp,
                 float* __restrict__ O) {
  __shared__ float sK[16][17];
  __shared__ float sV[16][17];
  __shared__ float sP[16][17];
  int qt = blockIdx.x, h = blockIdx.y, b = blockIdx.z;
  int lane = threadIdx.x;
  int lo = lane & 15;
  int hi = lane >> 4;

  const size_t base = (size_t)b * NPAD * DD + h * DKK;

  // preload Q fragments (A-matrix 16x4 layout, 4 k-steps over dk=16)
  v2f qa[4];
#pragma unroll
  for (int kk = 0; kk < 4; ++kk) {
    qa[kk][0] = Q[base + (size_t)(qt * 16 + lo) * DD + kk * 4 + 2 * hi + 0];
    qa[kk][1] = Q[base + (size_t)(qt * 16 + lo) * DD + kk * 4 + 2 * hi + 1];
  }

  float mi[8], li[8];
  v8f oacc;
#pragma unroll
  for (int r = 0; r < 8; ++r) { mi[r] = -1e30f; li[r] = 0.0f; oacc[r] = 0.0f; }

  for (int t = 0; t < NPAD / 16; ++t) {
    // stage K,V tiles (16 keys x 16 dk), coalesced
    {
      int row = lane >> 1, cb = (lane & 1) * 8;
      const float* kp = Kb + base + (size_t)(t * 16 + row) * DD + cb;
      const float* vpt = Vb + base + (size_t)(t * 16 + row) * DD + cb;
#pragma unroll
      for (int c = 0; c < 8; ++c) { sK[row][cb + c] = kp[c]; sV[row][cb + c] = vpt[c]; }
    }
    __syncthreads();

    // S = Q @ K^T
    v8f s;
#pragma unroll
    for (int r = 0; r < 8; ++r) s[r] = 0.0f;
#pragma unroll
    for (int kk = 0; kk < 4; ++kk) {
      v2f bf;  // B[k][n]: k = dk index, n = key index
      bf[0] = sK[lo][kk * 4 + 2 * hi + 0];
      bf[1] = sK[lo][kk * 4 + 2 * hi + 1];
      s = wmma4(qa[kk], bf, s);
    }

    int key = t * 16 + lo;  // this lane's column (key index)
    float pv[8];
#pragma unroll
    for (int r = 0; r < 8; ++r) {
      float sv = s[r] * 0.25f;           // 1/sqrt(16)
      if (key >= NTOK) sv = -1e30f;      // mask padded keys
      float rm = sv;
      rm = fmaxf(rm, __shfl_xor(rm, 1, 32));
      rm = fmaxf(rm, __shfl_xor(rm, 2, 32));
      rm = fmaxf(rm, __shfl_xor(rm, 4, 32));
      rm = fmaxf(rm, __shfl_xor(rm, 8, 32));
      float mn = fmaxf(mi[r], rm);
      float p = __expf(sv - mn);
      float rs = p;
      rs += __shfl_xor(rs, 1, 32);
      rs += __shfl_xor(rs, 2, 32);
      rs += __shfl_xor(rs, 4, 32);
      rs += __shfl_xor(rs, 8, 32);
      float sc = __expf(mi[r] - mn);
      li[r] = li[r] * sc + rs;
      mi[r] = mn;
      oacc[r] *= sc;
      pv[r] = p;
    }
    __syncthreads();
#pragma unroll
    for (int r = 0; r < 8; ++r) sP[r + 8 * hi][lo] = pv[r];
    __syncthreads();

    // O += P @ V
#pragma unroll
    for (int kk = 0; kk < 4; ++kk) {
      v2f pa, vb;
      pa[0] = sP[lo][kk * 4 + 2 * hi + 0];
      pa[1] = sP[lo][kk * 4 + 2 * hi + 1];
      vb[0] = sV[kk * 4 + 2 * hi + 0][lo];
      vb[1] = sV[kk * 4 + 2 * hi + 1][lo];
      oacc = wmma4(pa, vb, oacc);
    }
    __syncthreads();
  }

  float vpl = vp[b * DD + h * DKK + lo];
#pragma unroll
  for (int r = 0; r < 8; ++r) {
    int row = qt * 16 + r + 8 * hi;
    float o = oacc[r] / li[r];
    if (row < NTOK - 1) o += vpl;  // add broadcast cross-attn to node rows only
    O[(size_t)b * NPAD * DD + (size_t)row * DD + h * DKK + lo] = o;
  }
}

// ---------------------------------------------------------------- residual + instance norm
// Y[b,n,:] = IN(X[b,n,:] + T[b,n,:]) over n in [0,502); pad rows zeroed
__global__ void addnorm_kernel(const float* __restrict__ X, const float* __restrict__ T,
                               const float* __restrict__ g, const float* __restrict__ bt,
                               float* __restrict__ Y) {
  int b = blockIdx.x, d = threadIdx.x;
  const float* xp = X + (size_t)b * NPAD * DD + d;
  const float* tp = T + (size_t)b * NPAD * DD + d;
  float s = 0.0f, ss = 0.0f;
  for (int n = 0; n < NTOK; ++n) {
    float v = xp[(size_t)n * DD] + tp[(size_t)n * DD];
    s += v; ss += v * v;
  }
  float mu = s / (float)NTOK;
  float var = ss / (float)NTOK - mu * mu;
  float inv = rsqrtf(var + 1e-5f);
  float gg = g[d], bb = bt[d];
  float* yp = Y + (size_t)b * NPAD * DD + d;
  for (int n = 0; n < NTOK; ++n) {
    float v = xp[(size_t)n * DD] + tp[(size_t)n * DD];
    yp[(size_t)n * DD] = (v - mu) * inv * gg + bb;
  }
  for (int n = NTOK; n < NPAD; ++n) yp[(size_t)n * DD] = 0.0f;
}

// pack final output: rows 0..500 of each batch
__global__ void copyout_kernel(const float* __restrict__ X, float* __restrict__ out) {
  size_t i = (size_t)blockIdx.x * blockDim.x + threadIdx.x;
  const size_t total = (size_t)BB * (NNODE + 1) * DD;
  if (i < total) {
    int d = (int)(i % DD);
    size_t rest = i / DD;
    int n = (int)(rest % (NNODE + 1));
    int b = (int)(rest / (NNODE + 1));
    out[i] = X[((size_t)b * NPAD + n) * DD + d];
  }
}

// ---------------------------------------------------------------- launch
extern "C" void kernel_launch(void* const* d_in, const int* in_sizes, int n_in,
                              void* d_out, int out_size, void* d_ws, size_t ws_size,
                              hipStream_t stream) {
  (void)in_sizes; (void)n_in; (void)out_size; (void)ws_size;
  const float* depot = (const float*)d_in[0];
  const float* node  = (const float*)d_in[1];
  const float* pref  = (const float*)d_in[2];
  const float* Wp    = (const float*)d_in[3];
  const float* bp    = (const float*)d_in[4];
  const float* gd    = (const float*)d_in[5];
  const float* Wed   = (const float*)d_in[6];
  const float* bed   = (const float*)d_in[7];
  const float* gn    = (const float*)d_in[8];
  const float* Wen   = (const float*)d_in[9];
  const float* ben   = (const float*)d_in[10];
  const float* Wq1   = (const float*)d_in[11];
  const float* Wk1   = (const float*)d_in[12];
  const float* Wv1   = (const float*)d_in[13];
  // d_in[14]=Wq2, d_in[15]=Wk2 are dead (softmax over 1 key == 1)
  const float* Wv2   = (const float*)d_in[16];
  const float* Wc    = (const float*)d_in[17];
  const float* bc    = (const float*)d_in[18];
  const float* g1    = (const float*)d_in[19];
  const float* b1    = (const float*)d_in[20];
  const float* Wf1   = (const float*)d_in[21];
  const float* bf1   = (const float*)d_in[22];
  const float* Wf2   = (const float*)d_in[23];
  const float* bf2   = (const float*)d_in[24];
  const float* g2    = (const float*)d_in[25];
  const float* b2    = (const float*)d_in[26];

  float* ws = (float*)d_ws;
  const size_t SZ = (size_t)BB * NPAD * DD;       // 1,048,576 floats
  float* X   = ws;
  float* O1  = ws + 1 * SZ;
  float* Qb  = ws + 2 * SZ;                        // also reused as T1
  float* Kb  = ws + 3 * SZ;                        // also reused as T2
  float* Vb  = ws + 4 * SZ;
  float* Ob  = ws + 5 * SZ;
  float* H1  = ws + 6 * SZ;                        // B*512*512
  float* vp  = H1 + (size_t)BB * NPAD * FFF;
  float* st  = vp + BB * DD;                       // [imp_d, ld_d, imp_n, ld_n]

  float* out_main = (float*)d_out;
  float* out_loss = out_main + (size_t)BB * (NNODE + 1) * DD;
  float* out_ep   = out_loss + 1;

  // init
  zero_kernel<<<dim3((unsigned)((SZ + 255) / 256)), 256, 0, stream>>>(X, SZ);
  zero_kernel<<<1, 64, 0, stream>>>(st, 32);

  // embeddings
  ep_kernel<<<BB, DD, 0, stream>>>(pref, Wp, bp, X, out_ep);
  moe_kernel<<<BB, DD, 0, stream>>>(depot, gd, Wed, bed, X, st + 0, st + 8, 2, 1, 0);
  moe_kernel<<<BB * NNODE, DD, 0, stream>>>(node, gn, Wen, ben, X, st + 16, st + 24, 3, NNODE, 1);
  loss_kernel<<<1, 32, 0, stream>>>(st, out_loss);

  const int M = BB * NPAD;  // 8192
  dim3 gD(M / 128, DD / 64);   // N=128
  dim3 gF(M / 128, FFF / 64);  // N=512
  for (int l = 0; l < LLAY; ++l) {
    const size_t wo = (size_t)l * DD * DD;
    gemm_wmma<<<gD, 128, 0, stream>>>(X, Wq1 + wo, nullptr, Qb, M, DD, DD, 0);
    gemm_wmma<<<gD, 128, 0, stream>>>(X, Wk1 + wo, nullptr, Kb, M, DD, DD, 0);
    gemm_wmma<<<gD, 128, 0, stream>>>(X, Wv1 + wo, nullptr, Vb, M, DD, DD, 0);
    prefv_kernel<<<BB, DD, 0, stream>>>(X, Wv2 + wo, vp);
    attn_kernel<<<dim3(NPAD / 16, HH, BB), 32, 0, stream>>>(Qb, Kb, Vb, vp, Ob);
    gemm_wmma<<<gD, 128, 0, stream>>>(Ob, Wc + wo, bc + l * DD, Qb, M, DD, DD, 0);   // T1 = oc@Wc+bc
    addnorm_kernel<<<BB, DD, 0, stream>>>(X, Qb, g1 + l * DD, b1 + l * DD, O1);      // o1
    gemm_wmma<<<gF, 128, 0, stream>>>(O1, Wf1 + (size_t)l * DD * FFF, bf1 + l * FFF,
                                      H1, M, DD, FFF, 1);                            // gelu
    gemm_wmma<<<gD, 128, 0, stream>>>(H1, Wf2 + (size_t)l * FFF * DD, bf2 + l * DD,
                                      Kb, M, FFF, DD, 0);                            // T2 = o2
    addnorm_kernel<<<BB, DD, 0, stream>>>(O1, Kb, g2 + l * DD, b2 + l * DD, X);      // next X
  }

  const size_t outN = (size_t)BB * (NNODE + 1) * DD;
  copyout_kernel<<<dim3((unsigned)((outN + 255) / 256)), 256, 0, stream>>>(X, out_main);
}